// Graph_to_Featuremaps_savemem_6880537608450
// MI455X (gfx1250) — compile-verified
//
#include <hip/hip_runtime.h>

typedef __attribute__((ext_vector_type(2))) float v2f;
typedef __attribute__((ext_vector_type(8))) float v8f;

constexpr int kB = 16;     // batch
constexpr int kN = 7;      // nodes
constexpr int kH = 256;    // hidden
constexpr int kC = 256;    // channels
constexpr int kP = 9216;   // pixels = 96*96

// ---------------------------------------------------------------------------
// Kernel 1: per-batch b:
//   s_hid[n]  = sum_h x[b,n,h] * nfh[h]
//   attn[n]   = softmax_n(s_hid)          (s_res cancels: softmax shift-invariance)
//   xbar[b,h] = sum_n attn[n] * x[b,n,h]
// 16 blocks x 256 threads (8 waves/WGP).
// ---------------------------------------------------------------------------
__global__ void k_collapse(const float* __restrict__ x,    // (B, N, HID)
                           const float* __restrict__ nfh,  // (HID)
                           float* __restrict__ xbar) {     // (B, HID)
  __shared__ float red[kH];
  __shared__ float attn_s[kN];
  const int b = blockIdx.x;
  const int t = threadIdx.x;
  const float* xb = x + (size_t)b * kN * kH;
  const float nf = nfh[t];

  float xv[kN];
#pragma unroll
  for (int n = 0; n < kN; ++n) xv[n] = xb[n * kH + t];

  // 7 block-wide dot-product reductions (tiny; not on the critical path)
#pragma unroll
  for (int n = 0; n < kN; ++n) {
    red[t] = xv[n] * nf;
    __syncthreads();
    for (int off = kH / 2; off > 0; off >>= 1) {
      if (t < off) red[t] += red[t + off];
      __syncthreads();
    }
    if (t == 0) attn_s[n] = red[0];
    __syncthreads();
  }

  if (t == 0) {
    float s[kN];
    float m = -1e30f;
#pragma unroll
    for (int n = 0; n < kN; ++n) { s[n] = attn_s[n]; m = fmaxf(m, s[n]); }
    float sum = 0.f;
#pragma unroll
    for (int n = 0; n < kN; ++n) { s[n] = expf(s[n] - m); sum += s[n]; }
    const float inv = 1.f / sum;
#pragma unroll
    for (int n = 0; n < kN; ++n) attn_s[n] = s[n] * inv;
  }
  __syncthreads();

  float acc = 0.f;
#pragma unroll
  for (int n = 0; n < kN; ++n) acc = fmaf(attn_s[n], xv[n], acc);
  xbar[b * kH + t] = acc;
}

// ---------------------------------------------------------------------------
// Kernel 2: val(16x256) = relu( xbar(16x256) @ W(256x256) )
// V_WMMA_F32_16X16X4_F32: M=16 (one tile), 16 N-tiles (one wave each),
// K=256 in steps of 4; two interleaved accumulators halve the RAW chain.
// A layout (ISA 7.12.2, 32-bit A 16x4): lane m (lo half) holds K0,K1 in
// v0,v1; hi half-wave holds K2,K3. B/D: rows striped across lanes per VGPR.
// ---------------------------------------------------------------------------
__global__ void k_gemm_wmma(const float* __restrict__ xbar,  // (16, 256)  M x K
                            const float* __restrict__ w,     // (256, 256) K x N
                            float* __restrict__ val) {       // (16, 256)
  const int c0   = blockIdx.x * 16;
  const int lane = threadIdx.x;
  const int half = lane >> 4;   // 0: lanes 0-15, 1: lanes 16-31
  const int lm   = lane & 15;

  v8f acc0 = {};
  v8f acc1 = {};
  for (int k0 = 0; k0 < kH; k0 += 8) {
    const int ka = k0 + 2 * half;
    const int kb = ka + 4;
    v2f a0, b0, a1, b1;
    a0.x = xbar[lm * kH + ka];
    a0.y = xbar[lm * kH + ka + 1];
    b0.x = w[ka * kC + c0 + lm];
    b0.y = w[(ka + 1) * kC + c0 + lm];
    a1.x = xbar[lm * kH + kb];
    a1.y = xbar[lm * kH + kb + 1];
    b1.x = w[kb * kC + c0 + lm];
    b1.y = w[(kb + 1) * kC + c0 + lm];
    // (neg_a, A, neg_b, B, c_mod, C, reuse_a, reuse_b)
    acc0 = __builtin_amdgcn_wmma_f32_16x16x4_f32(false, a0, false, b0,
                                                 (short)0, acc0, false, false);
    acc1 = __builtin_amdgcn_wmma_f32_16x16x4_f32(false, a1, false, b1,
                                                 (short)0, acc1, false, false);
  }
  const v8f acc = acc0 + acc1;

#pragma unroll
  for (int r = 0; r < 8; ++r) {
    float v = acc[r];
    v = v > 0.f ? v : 0.f;                 // fused ReLU
    const int m = r + 8 * half;            // D layout: VGPR r -> row M
    val[m * kC + c0 + lm] = v;
  }
}

// ---------------------------------------------------------------------------
// Kernel 3: out[b,c,:] = val[b,c] broadcast over 9216 pixels.
// This is the entire runtime: 151 MB of b128 stores at the HBM write roofline
// (~6.5 us at 23.3 TB/s). 4096 blocks x 256 threads, 9 float4 stores/thread.
// ---------------------------------------------------------------------------
__global__ void k_bcast(const float* __restrict__ val,  // (B*C)
                        float* __restrict__ out) {      // (B*C, P)
  const int bc = blockIdx.x;
  const float v = val[bc];
  const float4 f = make_float4(v, v, v, v);
  float4* o = reinterpret_cast<float4*>(out) + (size_t)bc * (kP / 4);
  const int t = threadIdx.x;
#pragma unroll
  for (int i = 0; i < 9; ++i) o[t + i * 256] = f;  // 9*256 = 2304 = kP/4
}

// ---------------------------------------------------------------------------
extern "C" void kernel_launch(void* const* d_in, const int* in_sizes, int n_in,
                              void* d_out, int out_size, void* d_ws, size_t ws_size,
                              hipStream_t stream) {
  (void)in_sizes; (void)n_in; (void)out_size; (void)ws_size;
  // setup_inputs order: input, res_feature, node_fea_for_res, node_fea_for_hidden, weight
  const float* x   = (const float*)d_in[0];  // (1,B,N,HID)
  // d_in[1] (res_feature) and d_in[2] (node_fea_for_res) cancel in the softmax.
  const float* nfh = (const float*)d_in[3];  // (HID,1)
  const float* w   = (const float*)d_in[4];  // (HID,C)
  float* out = (float*)d_out;

  float* xbar = (float*)d_ws;          // 16*256 f32 = 16 KB
  float* val  = xbar + kB * kH;        // 16*256 f32 = 16 KB

  k_collapse<<<kB, kH, 0, stream>>>(x, nfh, xbar);
  k_gemm_wmma<<<kC / 16, 32, 0, stream>>>(xbar, w, val);
  k_bcast<<<kB * kC, 256, 0, stream>>>(val, out);
}